// Triplet_MMD_54176717472239
// MI455X (gfx1250) — compile-verified
//
#include <hip/hip_runtime.h>
#include <hip/hip_bf16.h>
#include <math.h>

typedef __attribute__((ext_vector_type(2))) float v2f;
typedef __attribute__((ext_vector_type(8))) float v8f;

#define NW_    8
#define STEP_  16
#define NPAIR  120
#define TT     512
#define DF     64
#define BIGF   1e10f
#define GAMMA_ 5.0f
#define INVG_  0.2f

// workspace layout in floats
#define O_NORMS 0          // 128*512 = 65536
#define O_DIST  65536      // 120 (padded to 128)
#define O_CROSS 65664      // 22*22*64 = 30976
#define O_L2    96640      // 30976
#define O_SCAL  127616     // [0]=sum(L2), [1]=mmd
#define O_D     131072     // D chunk buffer

// ---------------------------------------------------------------- utilities
__global__ void zero_kernel(float* scal) {
  if (threadIdx.x < 8) scal[threadIdx.x] = 0.0f;
}

// row squared-norms for all 128 sequences x 512 timesteps (one warp per row)
__global__ __launch_bounds__(256) void norms_kernel(const float* __restrict__ data,
                                                    float* __restrict__ norms) {
  int warp = (blockIdx.x * blockDim.x + threadIdx.x) >> 5;
  int lane = threadIdx.x & 31;
  if (warp >= 128 * TT) return;
  const float* row = data + (size_t)warp * DF;
  float a = row[lane], b = row[lane + 32];
  float s = a * a + b * b;
  #pragma unroll
  for (int m = 16; m; m >>= 1) s += __shfl_xor(s, m, 32);
  if (lane == 0) norms[warp] = s;
}

// ------------------------------------------------- pairwise-distance GEMM
// D[p,i,j] = |x_i|^2 + |y_j|^2 - 2 x_i . y_j   via V_WMMA_F32_16X16X4_F32
// block = 256 thr (8 waves), computes a 64(M) x 32(N) region; one 16x16
// tile per wave; K=64 swept in 16 steps of 4.
#define KPAD 68   // 64 + 4 pad: fragment reads land in distinct LDS banks

__global__ __launch_bounds__(256) void gemm_kernel(const float* __restrict__ data,
                                                   const float* __restrict__ norms,
                                                   float* __restrict__ Dbuf,
                                                   int pbase) {
  __shared__ float Xs[64 * KPAD];
  __shared__ float Ys[32 * KPAD];

  int p = pbase + blockIdx.y;
  int w = p / 15, r = p % 15;
  int aseq = w * STEP_;
  int oseq = w * STEP_ + 1 + r;

  int bm = blockIdx.x >> 4;      // 0..7  -> 64-row band
  int bn = blockIdx.x & 15;      // 0..15 -> 32-col band
  int gm0 = bm * 64, gn0 = bn * 32;
  int tid = threadIdx.x;

  const float* Xg = data + ((size_t)aseq * TT + gm0) * DF;
  const float* Yg = data + ((size_t)oseq * TT + gn0) * DF;

  // stage tiles (row-major, padded stride) -- coalesced b128 loads
  for (int idx = tid; idx < 64 * 16; idx += 256) {
    int row = idx >> 4, c4 = idx & 15;
    float4 v = ((const float4*)(Xg + row * DF))[c4];
    float* dst = &Xs[row * KPAD + c4 * 4];
    dst[0] = v.x; dst[1] = v.y; dst[2] = v.z; dst[3] = v.w;
  }
  for (int idx = tid; idx < 32 * 16; idx += 256) {
    int row = idx >> 4, c4 = idx & 15;
    float4 v = ((const float4*)(Yg + row * DF))[c4];
    float* dst = &Ys[row * KPAD + c4 * 4];
    dst[0] = v.x; dst[1] = v.y; dst[2] = v.z; dst[3] = v.w;
  }
  __syncthreads();

  int wave = tid >> 5, lane = tid & 31;
  int row0 = (wave & 3) * 16;    // 4 M-tiles
  int col0 = (wave >> 2) * 16;   // 2 N-tiles
  int mr = row0 + (lane & 15);
  int nr = col0 + (lane & 15);
  int kb = (lane >> 4) << 1;     // lanes 16-31 hold K+2,K+3 (ISA 16x4 layout)

  v8f acc = {0.f, 0.f, 0.f, 0.f, 0.f, 0.f, 0.f, 0.f};
  #pragma unroll
  for (int k = 0; k < 64; k += 4) {
    v2f a, b;
    a.x = Xs[mr * KPAD + k + kb];
    a.y = Xs[mr * KPAD + k + kb + 1];
    b.x = Ys[nr * KPAD + k + kb];
    b.y = Ys[nr * KPAD + k + kb + 1];
    acc = __builtin_amdgcn_wmma_f32_16x16x4_f32(false, a, false, b,
                                                (short)0, acc, false, false);
  }

  // C/D layout: VGPR v -> M = v (lanes 0-15) or v+8 (lanes 16-31); N = lane&15
  int colg = gn0 + col0 + (lane & 15);
  int rbase = gm0 + row0 + ((lane >> 4) << 3);
  float yn = norms[oseq * TT + colg];
  float* Dp = Dbuf + (size_t)blockIdx.y * (TT * TT);
  #pragma unroll
  for (int v = 0; v < 8; v++) {
    int rowg = rbase + v;
    float xn = norms[aseq * TT + rowg];
    Dp[(size_t)rowg * TT + colg] = xn + yn - 2.0f * acc[v];
  }
}

// ------------------------------------------------- soft-DTW anti-diagonal
__global__ __launch_bounds__(512) void dtw_kernel(const float* __restrict__ Dbuf,
                                                  const int* __restrict__ lens,
                                                  float* __restrict__ dist,
                                                  int pbase) {
  __shared__ float rb[3][TT];
  int pl = blockIdx.x;
  int p = pbase + pl;
  int w = p / 15, r = p % 15;
  int aseq = w * STEP_, oseq = aseq + 1 + r;
  int lx = lens[aseq], ly = lens[oseq];
  float invLen = 1.0f / (float)(lx + ly);
  const float* Dp = Dbuf + (size_t)pl * TT * TT;
  int i = threadIdx.x;

  for (int d = 0; d < 2 * TT - 1; d++) {
    int j = d - i;
    if (j >= 0 && j < TT) {
      int cur = d % 3, prv = (d + 2) % 3, pp = (d + 1) % 3;
      float left = (j >= 1) ? rb[prv][i] : BIGF;          // R[i][j-1]
      float up   = (i >= 1) ? rb[prv][i - 1] : BIGF;      // R[i-1][j]
      float dg;                                           // R[i-1][j-1]
      if (i >= 1 && j >= 1)      dg = rb[pp][i - 1];
      else if (i == 0 && j == 0) dg = 0.0f;
      else                       dg = BIGF;
      float m = fminf(left, fminf(up, dg));
      float s = __expf((m - left) * INVG_) + __expf((m - up) * INVG_) +
                __expf((m - dg) * INVG_);
      float rv = Dp[(size_t)i * TT + j] + (m - GAMMA_ * __logf(s));
      rb[cur][i] = rv;
      if (i == lx - 1 && j == ly - 1) dist[p] = rv * invLen;
    }
    __syncthreads();
  }
}

// ------------------------------------------------- MMD pieces
// cross[a,b,f] = sum_t total[a,t,f]*total[b,t,f];  total = data[0:11] ++ data[16:27]
__global__ void mmd_cross_kernel(const float* __restrict__ data,
                                 float* __restrict__ crossb) {
  int a = blockIdx.x / 22, b = blockIdx.x % 22;
  int f = threadIdx.x;  // 64 threads
  int sa = (a < 11) ? a : a + 5;
  int sb = (b < 11) ? b : b + 5;
  const float* pa = data + (size_t)sa * TT * DF + f;
  const float* pb = data + (size_t)sb * TT * DF + f;
  float s = 0.f;
  for (int t = 0; t < TT; t++) s += pa[(size_t)t * DF] * pb[(size_t)t * DF];
  crossb[(a * 22 + b) * 64 + f] = s;
}

// L2[a,b,f] = cross[a,a,f] + cross[b,b,f] - 2*cross[a,b,f];  sum -> scal[0]
__global__ __launch_bounds__(256) void mmd_l2_kernel(const float* __restrict__ crossb,
                                                     float* __restrict__ l2b,
                                                     float* __restrict__ scal) {
  int idx = blockIdx.x * blockDim.x + threadIdx.x;  // exactly 30976
  float l2 = 0.f;
  if (idx < 22 * 22 * 64) {
    int f = idx & 63;
    int ab = idx >> 6;
    int a = ab / 22, b = ab % 22;
    l2 = crossb[(a * 22 + a) * 64 + f] + crossb[(b * 22 + b) * 64 + f] -
         2.0f * crossb[idx];
    l2b[idx] = l2;
  }
  __shared__ float red[8];
  float s = l2;
  #pragma unroll
  for (int m = 16; m; m >>= 1) s += __shfl_xor(s, m, 32);
  int lane = threadIdx.x & 31, warp = threadIdx.x >> 5;
  if (lane == 0) red[warp] = s;
  __syncthreads();
  if (warp == 0) {
    s = (lane < 8) ? red[lane] : 0.f;
    #pragma unroll
    for (int m = 4; m; m >>= 1) s += __shfl_xor(s, m, 32);
    if (lane == 0) atomicAdd(&scal[0], s);
  }
}

// 5-bandwidth Gaussian kernel sum -> mmd (scal[1])
__global__ __launch_bounds__(256) void mmd_final_kernel(const float* __restrict__ l2b,
                                                        float* __restrict__ scal) {
  float bw = scal[0] * (1.0f / 462.0f) * 0.25f;  // /(n^2-n) / KMUL^(KNUM/2)
  float s = 0.f;
  for (int idx = threadIdx.x; idx < 11 * 11 * 64; idx += blockDim.x) {
    int f = idx & 63;
    int ab = idx >> 6;
    int a = ab / 11, b = ab % 11;
    float lxx = l2b[((a)      * 22 + b)      * 64 + f];
    float lyy = l2b[((a + 11) * 22 + b + 11) * 64 + f];
    float lxy = l2b[((a)      * 22 + b + 11) * 64 + f];
    float lyx = l2b[((a + 11) * 22 + b)      * 64 + f];
    float scale = 1.0f / bw;
    float t = 0.f;
    #pragma unroll
    for (int i = 0; i < 5; i++) {
      t += __expf(-lxx * scale) + __expf(-lyy * scale) -
           __expf(-lxy * scale) - __expf(-lyx * scale);
      scale *= 0.5f;
    }
    s += t;
  }
  __shared__ float red[8];
  #pragma unroll
  for (int m = 16; m; m >>= 1) s += __shfl_xor(s, m, 32);
  int lane = threadIdx.x & 31, warp = threadIdx.x >> 5;
  if (lane == 0) red[warp] = s;
  __syncthreads();
  if (warp == 0) {
    s = (lane < 8) ? red[lane] : 0.f;
    #pragma unroll
    for (int m = 4; m; m >>= 1) s += __shfl_xor(s, m, 32);
    if (lane == 0) scal[1] = s * (1.0f / 7744.0f);  // mean; ALPHA = 1.0
  }
}

// ------------------------------------------------- final loss
__global__ void loss_kernel(const float* __restrict__ dist,
                            const float* __restrict__ scal,
                            float* __restrict__ out) {
  __shared__ float dgs[40];
  __shared__ float wloss[8];
  int tid = threadIdx.x;  // 64 threads
  if (tid < 40) {
    int w = tid / 5, g = tid % 5;
    dgs[tid] = dist[w * 15 + g];
  }
  __syncthreads();
  if (tid < 8) {
    int w = tid;
    float dn[10];
    #pragma unroll
    for (int k = 0; k < 10; k++) dn[k] = dist[w * 15 + 5 + k];
    float sum = 0.f, sumdg = 0.f;
    int nz = 0;
    for (int g = 0; g < 5; g++) {
      float dgv = dgs[w * 5 + g];
      sumdg += dgv;
      #pragma unroll
      for (int k = 0; k < 10; k++) {
        float wgt = (k < 5) ? 0.7f : 0.3f;   // P_W / (1-P_W)
        float h = dgv + 1.0f - dn[k];        // MARGIN = 1.0
        if (h > 0.f) { sum += h * wgt; nz++; }
      }
    }
    float lv = sum / (float)(nz + 1);
    float only_pos = sumdg * (0.01f / 5.0f);  // LAM/NG
    wloss[w] = lv + only_pos;
  }
  __syncthreads();
  if (tid == 0) {
    float tl = 0.f;
    for (int w = 0; w < 8; w++) tl += wloss[w];
    tl *= (1.0f / 8.0f);
    float mu = 0.f;
    for (int i = 0; i < 40; i++) mu += dgs[i];
    mu *= (1.0f / 40.0f);
    float var = 0.f;
    for (int i = 0; i < 40; i++) { float d = dgs[i] - mu; var += d * d; }
    var *= (1.0f / 39.0f);                   // ddof=1
    out[0] = tl + scal[1] + var * 0.1f;      // + mmd*ALPHA + var_g*R_W
  }
}

// ------------------------------------------------- host
extern "C" void kernel_launch(void* const* d_in, const int* in_sizes, int n_in,
                              void* d_out, int out_size, void* d_ws, size_t ws_size,
                              hipStream_t stream) {
  (void)in_sizes; (void)n_in; (void)out_size;
  const float* data = (const float*)d_in[0];
  const int*   lens = (const int*)d_in[1];
  float* ws    = (float*)d_ws;
  float* norms = ws + O_NORMS;
  float* dist  = ws + O_DIST;
  float* crossb = ws + O_CROSS;
  float* l2b   = ws + O_L2;
  float* scal  = ws + O_SCAL;
  float* Dbuf  = ws + O_D;
  float* out   = (float*)d_out;

  zero_kernel<<<1, 32, 0, stream>>>(scal);
  norms_kernel<<<8192, 256, 0, stream>>>(data, norms);

  // MMD branch (independent, tiny)
  mmd_cross_kernel<<<484, 64, 0, stream>>>(data, crossb);
  mmd_l2_kernel<<<121, 256, 0, stream>>>(crossb, l2b, scal);
  mmd_final_kernel<<<1, 256, 0, stream>>>(l2b, scal);

  // Chunked GEMM -> DTW pipeline; chunk sized so D stays L2-resident
  long capFloats = (long)(ws_size / sizeof(float)) - O_D;
  int chunk = (int)(capFloats / ((long)TT * TT));
  if (chunk > 40) chunk = 40;   // 40 MB of D << 192 MB L2
  if (chunk < 1)  chunk = 1;
  for (int base = 0; base < NPAIR; base += chunk) {
    int np = NPAIR - base;
    if (np > chunk) np = chunk;
    gemm_kernel<<<dim3(128, np), 256, 0, stream>>>(data, norms, Dbuf, base);
    dtw_kernel<<<np, 512, 0, stream>>>(Dbuf, lens, dist, base);
  }

  loss_kernel<<<1, 64, 0, stream>>>(dist, scal, out);
}